// SpatialAggregationVectorEncoding_88527865905600
// MI455X (gfx1250) — compile-verified
//
#include <hip/hip_runtime.h>
#include <hip/hip_bf16.h>

// ---------------------------------------------------------------------------
// SpatialAggregationVectorEncoding for MI455X (gfx1250), wave32 + WMMA + TDM.
//
// out[b,h,m,d] = x[b,h,m,d] + sum_{n,k} table[m,n,k] * P[k,d,h] * x[b,h,n,d],
// P = param_1 + param_2. The table is separable over the 14x14 token grid:
//   table[(r,c),(r',c'),k] = delta_{c,c'}*V[r,r',k] + delta_{r,r'}*Hh[c,c',k]
// Folding P per (h,d) gives tiny L2-resident matrices Wv/Wh (16x16, zero
// padded). Per (b,h,d), with Xg[r,c] = x[b,h,(r*14+c),d]:
//   out = Xg + Wv @ Xg + Xg @ Wh^T
// -> two 14x14x14 f32 matmuls -> 4+4 chained V_WMMA_F32_16X16X4_F32,
// two independent accumulator chains per wave for ILP.
// Memory bound (~40 MB traffic vs ~0.3 GFLOP), so full-f32 WMMA is free.
//
// Data movement:
//  - x slab (196x64 f32, contiguous) staged to LDS by the Tensor Data Mover:
//    token-major layout Xs[tok*65 + d]; the +1 dword per 64 is produced by
//    the TDM's pad feature (pad_interval=64 dwords, pad_amount=1 dword).
//  - OOB fragment reads are redirected to a zeroed LDS page so the validity
//    select is loop-invariant (hoisted out of the d-loop entirely).
//  - results written back into LDS, then copied out as coalesced b128.
// ---------------------------------------------------------------------------

typedef __attribute__((ext_vector_type(2))) float v2f;
typedef __attribute__((ext_vector_type(8))) float v8f;
typedef __attribute__((ext_vector_type(4))) unsigned int v4u;
typedef __attribute__((ext_vector_type(4))) int v4i;
typedef __attribute__((ext_vector_type(8))) int v8i;

#define GH 14
#define GW 14
#define L_TOK 196
#define NNODE 16
#define NB 32
#define NH 12
#define NDH 64
#define W_SLAB 256              // 16*16 floats per (h,d) weight matrix
#define TOK_S 65                // LDS token stride: 64 data + 1 pad dword
#define ZBASE (L_TOK * TOK_S)   // 12740: zero page base
#define ZPAGE 132               // covers ZBASE + 65 + 63 reads
#define XS_FLOATS (ZBASE + ZPAGE)

#if defined(__has_builtin)
#if __has_builtin(__builtin_amdgcn_tensor_load_to_lds) && \
    __has_builtin(__builtin_amdgcn_s_wait_tensorcnt)
#define HAVE_TDM 1
#endif
#endif

// ---------------------------------------------------------------------------
// Kernel A: build Wv[h][d][16][16], Wh[h][d][16][16] (zero padded) from
// table + (param_1 + param_2). One thread per element.
//   V[r,r',k]  = table[(r*14+0 ), (r'*14+0 ), k], k in 0..7
//   Hh[c,c',k] = table[(0*14+c ), (0*14+c'), k], k in 8..15
// ---------------------------------------------------------------------------
__global__ void build_weights_kernel(const float* __restrict__ table,
                                     const float* __restrict__ p1,
                                     const float* __restrict__ p2,
                                     float* __restrict__ Wv,
                                     float* __restrict__ Wh) {
  const int idx = blockIdx.x * blockDim.x + threadIdx.x;   // 12*64*16*16
  const int rp = idx & 15;
  const int r  = (idx >> 4) & 15;
  const int d  = (idx >> 8) & 63;
  const int h  = idx >> 14;
  if (h >= NH) return;

  float wv = 0.0f, wh = 0.0f;
  if (r < GH && rp < GH) {
    const float* tv = table + ((size_t)(r * GW) * L_TOK + (size_t)(rp * GW)) * NNODE;
    const float* th = table + ((size_t)r * L_TOK + (size_t)rp) * NNODE;
#pragma unroll
    for (int k = 0; k < 8; ++k) {
      const float pv = p1[(size_t)k * NDH * NH + d * NH + h] +
                       p2[(size_t)k * NDH * NH + d * NH + h];
      const float ph = p1[(size_t)(k + 8) * NDH * NH + d * NH + h] +
                       p2[(size_t)(k + 8) * NDH * NH + d * NH + h];
      wv += tv[k] * pv;
      wh += th[k + 8] * ph;
    }
  }
  const size_t o = ((size_t)(h * NDH + d) * 16 + r) * 16 + rp;
  Wv[o] = wv;
  Wh[o] = wh;
}

// ---------------------------------------------------------------------------
// Kernel B: one workgroup (256 threads = 8 wave32) per (b,h).
// Fragment layouts (ISA 7.12.2, f32; half = lane>=16, lid = lane&15):
//   A (16x4): M = lid; VGPR0 = A[M, k0], VGPR1 = A[M, k0+1], k0 = kb+2*half
//   B (4x16): N = lid; VGPR0 = B[k0, N], VGPR1 = B[k0+1, N] (mirror of A)
//   C (16x16): VGPR v -> row M = v + 8*half, col N = lid
// ---------------------------------------------------------------------------
__global__ __launch_bounds__(256) void spatial_agg_kernel(
    const float* __restrict__ x,
    const float* __restrict__ Wv,
    const float* __restrict__ Wh,
    float* __restrict__ out) {
  __shared__ float Xs[XS_FLOATS];

  const int bh = blockIdx.x;           // 0..383
  const int h  = bh % NH;
  const int tid = threadIdx.x;
  const int wave = tid >> 5;
  const int lane = tid & 31;
  const int half = lane >> 4;
  const int lid  = lane & 15;

  const float* xg = x + (size_t)bh * L_TOK * NDH;
  float* og = out + (size_t)bh * L_TOK * NDH;
  const float* wv_head = Wv + (size_t)h * NDH * W_SLAB;
  const float* wh_head = Wh + (size_t)h * NDH * W_SLAB;

  // Warm the L2-resident per-head weight slabs toward the WGP.
  __builtin_prefetch(wv_head + tid * 16, 0, 0);
  __builtin_prefetch(wh_head + tid * 16, 0, 0);

  // Zero page for OOB fragment reads.
  if (tid < ZPAGE) Xs[ZBASE + tid] = 0.0f;

#if HAVE_TDM
  if (wave == 0) {
    // ---- Tensor Data Mover: 12544 contiguous dwords -> LDS with a 1-dword
    // pad after every 64 dwords (token stride 65). D# per ISA ch.8. ----
    const unsigned lds_addr = (unsigned)(uintptr_t)(&Xs[0]);
    const unsigned long long ga = (unsigned long long)(uintptr_t)xg;
    v4u g0;
    g0[0] = 1u;                                  // count=1, user mode
    g0[1] = lds_addr;                            // lds_addr (bytes)
    g0[2] = (unsigned)(ga & 0xffffffffu);        // global_addr[31:0]
    g0[3] = (unsigned)((ga >> 32) & 0x01ffffffu) // global_addr[56:32]
            | 0x80000000u;                       // type=2 ("image")
    v8i g1;
    g1[0] = (int)((2u << 16)      // data_size = 4 bytes
                | (1u << 20)      // pad_enable
                | (5u << 22));    // pad_interval: 2^(5+1) = 64 dwords
                                  // pad_amount encoded 0 -> 1 dword
    g1[1] = (int)((unsigned)(L_TOK * NDH) << 16);  // tensor_dim0 lo16 @bit48
    g1[2] = (int)(((unsigned)(L_TOK * NDH) >> 16)  // tensor_dim0 hi16
                | (1u << 16));                     // tensor_dim1 = 1
    g1[3] = (int)((unsigned)(L_TOK * NDH) << 16);  // tile_dim0 @bit112
    g1[4] = 0;                                     // tile_dim1/2 unused
    g1[5] = (int)(L_TOK * NDH);                    // tensor_dim0_stride lo32
    g1[6] = 0;
    g1[7] = 0;
    v4i gz = {0, 0, 0, 0};
#if __clang_major__ >= 23
    v8i gz8 = {0, 0, 0, 0, 0, 0, 0, 0};
    __builtin_amdgcn_tensor_load_to_lds(g0, g1, gz, gz, gz8, 0);
#else
    __builtin_amdgcn_tensor_load_to_lds(g0, g1, gz, gz, 0);
#endif
    __builtin_amdgcn_s_wait_tensorcnt(0);
  }
#else
  // Fallback: coalesced b128 loads, scalar LDS scatter into stride-65 layout.
  for (int i = tid; i < (L_TOK * NDH) / 4; i += 256) {
    const float4 vl = ((const float4*)xg)[i];
    const int tok = i >> 4;
    const int base = tok * TOK_S + ((i & 15) << 2);
    Xs[base + 0] = vl.x;
    Xs[base + 1] = vl.y;
    Xs[base + 2] = vl.z;
    Xs[base + 3] = vl.w;
  }
#endif
  __syncthreads();

  // ---- Loop-invariant fragment base offsets (validity folded into the
  // base: invalid -> ZBASE, whose whole +0..+128 range is zeroed). For even
  // k0, columns k0 and k0+1 share validity, so the horizontal pair shares
  // one base (addresses +0 / +65 -> single ds_load_2addr_b32). ----
  int vbx[4], vby[4], hb[4];
#pragma unroll
  for (int kt = 0; kt < 4; ++kt) {
    const int k0 = 4 * kt + 2 * half;
    const bool vok = (k0 + 1 < GH) && (lid < GW);   // k0 even: rows k0,k0+1 同 validity
    vbx[kt] = vok ? (k0 * GW + lid) * TOK_S : ZBASE;
    vby[kt] = vok ? ((k0 + 1) * GW + lid) * TOK_S : ZBASE;
    const bool hok = (lid < GH) && (k0 + 1 < GW);   // cols k0,k0+1 share validity
    hb[kt] = hok ? (lid * GW + k0) * TOK_S : ZBASE;
  }

  // Each wave: channels {wave, wave+16, wave+32, wave+48} paired with +8.
  for (int d0 = wave; d0 < NDH; d0 += 16) {
    const int dch[2] = {d0, d0 + 8};
    v8f acc[2] = {{}, {}};

#pragma unroll
    for (int kt = 0; kt < 4; ++kt) {
      const int k0 = 4 * kt + 2 * half;
      // ---- vertical: acc += Wv @ Xg ----
#pragma unroll
      for (int j = 0; j < 2; ++j) {
        v2f a = *(const v2f*)(wv_head + dch[j] * W_SLAB + lid * 16 + k0);
        v2f b;
        b.x = Xs[vbx[kt] + dch[j]];
        b.y = Xs[vby[kt] + dch[j]];
        acc[j] = __builtin_amdgcn_wmma_f32_16x16x4_f32(false, a, false, b,
                                                       (short)0, acc[j], false, false);
      }
    }

#pragma unroll
    for (int kt = 0; kt < 4; ++kt) {
      const int k0 = 4 * kt + 2 * half;
      // ---- horizontal: acc += Xg @ Wh^T ----
#pragma unroll
      for (int j = 0; j < 2; ++j) {
        v2f a;
        a.x = Xs[hb[kt] + dch[j]];
        a.y = Xs[hb[kt] + TOK_S + dch[j]];       // same base -> ds_load_2addr
        v2f b = *(const v2f*)(wh_head + dch[j] * W_SLAB + lid * 16 + k0);
        acc[j] = __builtin_amdgcn_wmma_f32_16x16x4_f32(false, a, false, b,
                                                       (short)0, acc[j], false, false);
      }
    }

    // ---- identity + writeback into LDS (unique cell per lane) ----
#pragma unroll
    for (int j = 0; j < 2; ++j) {
#pragma unroll
      for (int v = 0; v < 8; ++v) {
        const int r = v + 8 * half;
        if (r < GH && lid < GW) {
          const int idx = (r * GW + lid) * TOK_S + dch[j];
          Xs[idx] = Xs[idx] + acc[j][v];
        }
      }
    }
  }
  __syncthreads();

  // ---- coalesced b128 copy-out: each wave stores 512B contiguous ----
  for (int i = tid; i < (L_TOK * NDH) / 4; i += 256) {
    const int tok = i >> 4;
    const int base = tok * TOK_S + ((i & 15) << 2);
    float4 o;
    o.x = Xs[base + 0];
    o.y = Xs[base + 1];
    o.z = Xs[base + 2];
    o.w = Xs[base + 3];
    ((float4*)og)[i] = o;
  }
}

extern "C" void kernel_launch(void* const* d_in, const int* in_sizes, int n_in,
                              void* d_out, int out_size, void* d_ws, size_t ws_size,
                              hipStream_t stream) {
  const float* x     = (const float*)d_in[0];  // [32,12,196,64]
  const float* table = (const float*)d_in[1];  // [196,196,16]
  const float* p1    = (const float*)d_in[2];  // [16,64,12]
  const float* p2    = (const float*)d_in[3];  // [16,64,12]
  float* out = (float*)d_out;

  // Workspace: Wv + Wh, each 12*64*256 floats = 786 KB (1.57 MB total).
  float* Wv = (float*)d_ws;
  float* Wh = Wv + (size_t)NH * NDH * W_SLAB;

  {
    const int total = NH * NDH * 16 * 16;      // 196608
    build_weights_kernel<<<total / 256, 256, 0, stream>>>(table, p1, p2, Wv, Wh);
  }
  spatial_agg_kernel<<<NB * NH, 256, 0, stream>>>(x, Wv, Wh, out);
}